// SimpleEncoder_43611097924235
// MI455X (gfx1250) — compile-verified
//
#include <hip/hip_runtime.h>

// ---------------------------------------------------------------------------
// CDNA5 (gfx1250) implementation of the SimpleEncoder reference:
//   - two 3-layer fp32 MLPs via V_WMMA_F32_16X16X4_F32 (full fp32 matrix pipe)
//   - degree-normalized bidirectional SpMM x3 via L2-resident f32 atomics
// ---------------------------------------------------------------------------

typedef __attribute__((ext_vector_type(2))) float v2f;
typedef __attribute__((ext_vector_type(8))) float v8f;

#define HID_LD 132  // padded LDS row stride (floats) to spread banks

__device__ __forceinline__ void atomAddF(float* p, float v) {
  // hardware global_atomic_add_f32 (no CAS loop)
  unsafeAtomicAdd(p, v);
}

// ---------------------------------------------------------------------------
// Fused 3-layer MLP: X(N x 256) -> relu -> 128 -> relu -> 128 -> 64
// One block processes 32 rows through all layers. 8 waves per block.
// Wave tile = 16x16 output; WMMA f32 16x16x4, K stepped by 4.
// A layout (ISA 7.12.2, 32-bit A 16x4): lane l: M = l&15, VGPR r holds
// K = k + 2*(l>>4) + r  -> contiguous float2 per lane.
// B layout (4x16): VGPR r: K = k + 2*(l>>4) + r, N = l&15.
// C/D layout: VGPR v: M = v + 8*(l>>4), N = l&15.
// ---------------------------------------------------------------------------
__global__ __launch_bounds__(256) void mlp3_kernel(
    const float* __restrict__ Xs, const float* __restrict__ Xt,
    const float* __restrict__ Ws0, const float* __restrict__ bs0,
    const float* __restrict__ Wt0, const float* __restrict__ bt0,
    const float* __restrict__ Ws1, const float* __restrict__ bs1,
    const float* __restrict__ Wt1, const float* __restrict__ bt1,
    const float* __restrict__ Ws2, const float* __restrict__ bs2,
    const float* __restrict__ Wt2, const float* __restrict__ bt2,
    float* __restrict__ outS, float* __restrict__ outT, int nrows)
{
  __shared__ float h1[32 * HID_LD];
  __shared__ float h2[32 * HID_LD];

  const bool isT = (blockIdx.y != 0);
  const float* X  = isT ? Xt  : Xs;
  const float* W0 = isT ? Wt0 : Ws0;  const float* B0 = isT ? bt0 : bs0;
  const float* W1 = isT ? Wt1 : Ws1;  const float* B1 = isT ? bt1 : bs1;
  const float* W2 = isT ? Wt2 : Ws2;  const float* B2 = isT ? bt2 : bs2;
  float* OUT = isT ? outT : outS;

  const int lane = (int)(threadIdx.x & 31u);
  const int wave = (int)(threadIdx.x >> 5u);
  const int li   = lane & 15;
  const int grp  = lane >> 4;
  const int rowBlock = (int)blockIdx.x * 32;

  // ---------------- layer 1: X(32x256) @ W0(256x128), relu -> h1 ----------
  #pragma unroll
  for (int t = 0; t < 2; ++t) {
    const int tile = wave + 8 * t;        // 16 tiles: 2 rowgroups x 8 coltiles
    const int rg = tile >> 3, ct = tile & 7;
    const int n0 = ct * 16;
    int arowIdx = rowBlock + rg * 16 + li;
    if (arowIdx >= nrows) arowIdx = nrows - 1;   // clamp (N % 32 == 0 anyway)
    const float* arow = X + (size_t)arowIdx * 256 + 2 * grp;
    v8f acc = {};
    #pragma unroll 8
    for (int k = 0; k < 256; k += 4) {
      v2f a = *(const v2f*)(arow + k);
      const float* bp = W0 + (size_t)(k + 2 * grp) * 128 + n0 + li;
      v2f b = { bp[0], bp[128] };
      acc = __builtin_amdgcn_wmma_f32_16x16x4_f32(false, a, false, b,
                                                  (short)0, acc, false, false);
    }
    const float bias = B0[n0 + li];
    float* hrow = h1 + (rg * 16 + 8 * grp) * HID_LD + n0 + li;
    #pragma unroll
    for (int v = 0; v < 8; ++v) {
      float x = acc[v] + bias;
      hrow[v * HID_LD] = x > 0.f ? x : 0.f;
    }
  }
  __syncthreads();

  // ---------------- layer 2: h1(32x128) @ W1(128x128), relu -> h2 ---------
  #pragma unroll
  for (int t = 0; t < 2; ++t) {
    const int tile = wave + 8 * t;
    const int rg = tile >> 3, ct = tile & 7;
    const int n0 = ct * 16;
    const float* arow = h1 + (rg * 16 + li) * HID_LD + 2 * grp;
    v8f acc = {};
    #pragma unroll 8
    for (int k = 0; k < 128; k += 4) {
      v2f a = *(const v2f*)(arow + k);
      const float* bp = W1 + (size_t)(k + 2 * grp) * 128 + n0 + li;
      v2f b = { bp[0], bp[128] };
      acc = __builtin_amdgcn_wmma_f32_16x16x4_f32(false, a, false, b,
                                                  (short)0, acc, false, false);
    }
    const float bias = B1[n0 + li];
    float* hrow = h2 + (rg * 16 + 8 * grp) * HID_LD + n0 + li;
    #pragma unroll
    for (int v = 0; v < 8; ++v) {
      float x = acc[v] + bias;
      hrow[v * HID_LD] = x > 0.f ? x : 0.f;
    }
  }
  __syncthreads();

  // ---------------- layer 3: h2(32x128) @ W2(128x64) -> OUT (no relu) -----
  {
    const int rg = wave >> 2, ct = wave & 3;   // 8 tiles: 2 rg x 4 coltiles
    const int n0 = ct * 16;
    const float* arow = h2 + (rg * 16 + li) * HID_LD + 2 * grp;
    v8f acc = {};
    #pragma unroll 8
    for (int k = 0; k < 128; k += 4) {
      v2f a = *(const v2f*)(arow + k);
      const float* bp = W2 + (size_t)(k + 2 * grp) * 64 + n0 + li;
      v2f b = { bp[0], bp[64] };
      acc = __builtin_amdgcn_wmma_f32_16x16x4_f32(false, a, false, b,
                                                  (short)0, acc, false, false);
    }
    const float bias = B2[n0 + li];
    const int rowBase = rowBlock + rg * 16 + 8 * grp;
    #pragma unroll
    for (int v = 0; v < 8; ++v) {
      const int r = rowBase + v;
      if (r < nrows) OUT[(size_t)r * 64 + n0 + li] = acc[v] + bias;
    }
  }
}

// ---------------------------------------------------------------------------
// Degree kernels (self-loops folded in via init value of 1.0)
// ---------------------------------------------------------------------------
__global__ void deg_init_kernel(float* oi, float* ii, int n) {
  int i = (int)(blockIdx.x * blockDim.x + threadIdx.x);
  if (i < n) { oi[i] = 1.0f; ii[i] = 1.0f; }
}

__global__ void deg_acc_kernel(const int* __restrict__ row,
                               const int* __restrict__ col,
                               float* oi, float* ii, int e) {
  int i = (int)(blockIdx.x * blockDim.x + threadIdx.x);
  if (i < e) {
    atomAddF(&oi[row[i]], 1.0f);   // out-degree
    atomAddF(&ii[col[i]], 1.0f);   // in-degree
  }
}

__global__ void deg_inv_kernel(float* oi, float* ii, int n) {
  int i = (int)(blockIdx.x * blockDim.x + threadIdx.x);
  if (i < n) {
    float a = oi[i]; oi[i] = a > 0.f ? rsqrtf(a) : 0.f;
    float b = ii[i]; ii[i] = b > 0.f ? rsqrtf(b) : 0.f;
  }
}

// ---------------------------------------------------------------------------
// Fused ping-pong init + self-loop edge:
//   dst_s[n] = src_s[n] + c0 * w_n * src_t[n]
//   dst_t[n] = src_t[n] + c1 * w_n * src_s[n]
// One thread per (node, 4 features).
// ---------------------------------------------------------------------------
__global__ void selfloop_kernel(const float* __restrict__ ss,
                                const float* __restrict__ st,
                                float* __restrict__ ds, float* __restrict__ dt,
                                const float* __restrict__ oi,
                                const float* __restrict__ ii,
                                const float* __restrict__ cw, int k, int n) {
  int tid = (int)(blockIdx.x * blockDim.x + threadIdx.x);
  int node = tid >> 4;
  if (node >= n) return;
  int f = (tid & 15) * 4;
  const float c0 = cw[2 * k], c1 = cw[2 * k + 1];
  const float w = oi[node] * ii[node];
  const float a0 = c0 * w, a1 = c1 * w;
  const size_t base = (size_t)node * 64 + f;
  float4 sv = *(const float4*)(ss + base);
  float4 tv = *(const float4*)(st + base);
  float4 o1 = make_float4(sv.x + a0 * tv.x, sv.y + a0 * tv.y,
                          sv.z + a0 * tv.z, sv.w + a0 * tv.w);
  float4 o2 = make_float4(tv.x + a1 * sv.x, tv.y + a1 * sv.y,
                          tv.z + a1 * sv.z, tv.w + a1 * sv.w);
  *(float4*)(ds + base) = o1;
  *(float4*)(dt + base) = o2;
}

// ---------------------------------------------------------------------------
// Bidirectional edge scatter:
//   dst_s[r] += c0 * w_e * src_t[c]   (adj_norm  @ t)
//   dst_t[c] += c1 * w_e * src_s[r]   (adj_norm^T @ s)
// 8 threads per edge, 8 features each. Feature tables are L2-resident.
// ---------------------------------------------------------------------------
__global__ void edge_prop_kernel(const int* __restrict__ row,
                                 const int* __restrict__ col,
                                 const float* __restrict__ oi,
                                 const float* __restrict__ ii,
                                 const float* __restrict__ ss,
                                 const float* __restrict__ st,
                                 float* __restrict__ ds, float* __restrict__ dt,
                                 const float* __restrict__ cw, int k, int e) {
  int tid = (int)(blockIdx.x * blockDim.x + threadIdx.x);
  int eid = tid >> 3;
  if (eid >= e) return;
  int f = (tid & 7) * 8;
  const int r = row[eid], c = col[eid];
  const float w = oi[r] * ii[c];
  const float a0 = cw[2 * k] * w;
  const float a1 = cw[2 * k + 1] * w;
  const size_t sb = (size_t)r * 64 + f;
  const size_t tb = (size_t)c * 64 + f;
  const float4* tp = (const float4*)(st + tb);
  const float4* sp = (const float4*)(ss + sb);
  float4 t0 = tp[0], t1 = tp[1];
  float4 s0 = sp[0], s1 = sp[1];
  float* dsp = ds + sb;
  float* dtp = dt + tb;
  atomAddF(dsp + 0, a0 * t0.x); atomAddF(dsp + 1, a0 * t0.y);
  atomAddF(dsp + 2, a0 * t0.z); atomAddF(dsp + 3, a0 * t0.w);
  atomAddF(dsp + 4, a0 * t1.x); atomAddF(dsp + 5, a0 * t1.y);
  atomAddF(dsp + 6, a0 * t1.z); atomAddF(dsp + 7, a0 * t1.w);
  atomAddF(dtp + 0, a1 * s0.x); atomAddF(dtp + 1, a1 * s0.y);
  atomAddF(dtp + 2, a1 * s0.z); atomAddF(dtp + 3, a1 * s0.w);
  atomAddF(dtp + 4, a1 * s1.x); atomAddF(dtp + 5, a1 * s1.y);
  atomAddF(dtp + 6, a1 * s1.z); atomAddF(dtp + 7, a1 * s1.w);
}

// ---------------------------------------------------------------------------
// Launch. Input order (setup_inputs):
//  0 s, 1 t, 2 edge_index, 3 Ws0, 4 bs0, 5 Wt0, 6 bt0, 7 Ws1, 8 bs1,
//  9 Wt1, 10 bt1, 11 Ws2, 12 bs2, 13 Wt2, 14 bt2, 15 conv_w
// Output: s (N*64) then t (N*64), flat fp32.
// Workspace: s0,t0 (2 * N*64 f32) + out_inv,in_inv (2 * N f32) ~= 52 MB.
// ---------------------------------------------------------------------------
extern "C" void kernel_launch(void* const* d_in, const int* in_sizes, int n_in,
                              void* d_out, int out_size, void* d_ws,
                              size_t ws_size, hipStream_t stream) {
  const float* Xs  = (const float*)d_in[0];
  const float* Xt  = (const float*)d_in[1];
  const int*   ei  = (const int*)d_in[2];
  const float* Ws0 = (const float*)d_in[3];
  const float* bs0 = (const float*)d_in[4];
  const float* Wt0 = (const float*)d_in[5];
  const float* bt0 = (const float*)d_in[6];
  const float* Ws1 = (const float*)d_in[7];
  const float* bs1 = (const float*)d_in[8];
  const float* Wt1 = (const float*)d_in[9];
  const float* bt1 = (const float*)d_in[10];
  const float* Ws2 = (const float*)d_in[11];
  const float* bs2 = (const float*)d_in[12];
  const float* Wt2 = (const float*)d_in[13];
  const float* bt2 = (const float*)d_in[14];
  const float* cw  = (const float*)d_in[15];

  const int N = in_sizes[0] / 256;
  const int E = in_sizes[2] / 2;
  const int* rowp = ei;        // edge_index[0]
  const int* colp = ei + E;    // edge_index[1]

  float* s0 = (float*)d_ws;                  // MLP output / ping-pong pair P
  float* t0 = s0 + (size_t)N * 64;
  float* oi = t0 + (size_t)N * 64;           // out_inv (also out_deg scratch)
  float* ii = oi + N;                        // in_inv  (also in_deg scratch)
  float* sQ = (float*)d_out;                 // ping-pong pair Q = final out
  float* tQ = sQ + (size_t)N * 64;

  const int TB = 256;

  // MLPs for s and t (gridDim.y selects the weight set)
  dim3 mgrid((unsigned)((N + 31) / 32), 2u, 1u);
  mlp3_kernel<<<mgrid, TB, 0, stream>>>(Xs, Xt, Ws0, bs0, Wt0, bt0,
                                        Ws1, bs1, Wt1, bt1,
                                        Ws2, bs2, Wt2, bt2, s0, t0, N);

  // degrees -> deg^-1/2
  deg_init_kernel<<<(N + TB - 1) / TB, TB, 0, stream>>>(oi, ii, N);
  deg_acc_kernel<<<(E + TB - 1) / TB, TB, 0, stream>>>(rowp, colp, oi, ii, E);
  deg_inv_kernel<<<(N + TB - 1) / TB, TB, 0, stream>>>(oi, ii, N);

  // 3 propagation rounds, ping-pong P -> Q -> P -> Q(=d_out)
  float* srcS = s0; float* srcT = t0;
  float* dstS = sQ; float* dstT = tQ;
  const int slBlocks = (int)(((size_t)N * 16 + TB - 1) / TB);
  const int epBlocks = (int)(((size_t)E * 8 + TB - 1) / TB);
  for (int k = 0; k < 3; ++k) {
    selfloop_kernel<<<slBlocks, TB, 0, stream>>>(srcS, srcT, dstS, dstT,
                                                 oi, ii, cw, k, N);
    edge_prop_kernel<<<epBlocks, TB, 0, stream>>>(rowp, colp, oi, ii,
                                                  srcS, srcT, dstS, dstT,
                                                  cw, k, E);
    float* ns = dstS; float* nt = dstT;
    dstS = srcS; dstT = srcT;
    srcS = ns;  srcT = nt;
  }
}